// Net_65335042507316
// MI455X (gfx1250) — compile-verified
//
#include <hip/hip_runtime.h>

typedef float v2f __attribute__((ext_vector_type(2)));
typedef float v8f __attribute__((ext_vector_type(8)));

#define LDU 512            // padded hidden width (500 -> 512)
#define LDU4 16            // padded output width (10 -> 16)
#define EPS 0.5f

// ---------------------------------------------------------------------------
// fp32 WMMA: D = A(16x4) * B(4x16) + C  (V_WMMA_F32_16X16X4_F32)
// ---------------------------------------------------------------------------
__device__ __forceinline__ v8f wmma4(v2f a, v2f b, v8f c) {
  return __builtin_amdgcn_wmma_f32_16x16x4_f32(
      /*neg_a=*/false, a, /*neg_b=*/false, b,
      /*c_mod=*/(short)0, c, /*reuse_a=*/false, /*reuse_b=*/false);
}

// Swizzled-B GEMM: B stored in WMMA fragment order:
//   Bsw[ nt*tileStride + kc*64 + lane*2 + j ] = B[kc*4 + 2*(lane>>4) + j][nt*16 + (lane&15)]
// so every fragment is one coalesced b64 load per lane.
// Accumulates a 16x32 strip (two adjacent n-tiles nt0, nt0+1).
__device__ __forceinline__ void gemm2s(const float* __restrict__ A, int lda,
                                       const float* __restrict__ Bsw, int nt0,
                                       int tileStride, int K,
                                       int am, int lane, v8f& acc0, v8f& acc1) {
  const int hi = lane >> 4;
  const float* arow = A + am * lda + 2 * hi;
  const float* bp0 = Bsw + nt0 * tileStride + lane * 2;
  const float* bp1 = bp0 + tileStride;
#pragma unroll 4
  for (int k0 = 0; k0 < K; k0 += 4) {
    v2f a  = *(const v2f*)(arow + k0);
    v2f b0 = *(const v2f*)(bp0 + (k0 << 4));   // (k0/4)*64
    v2f b1 = *(const v2f*)(bp1 + (k0 << 4));
    acc0 = wmma4(a, b0, acc0);
    acc1 = wmma4(a, b1, acc1);
  }
}

// Single-tile variant (16x16 strip, n-tile 0)
__device__ __forceinline__ void gemm1s(const float* __restrict__ A, int lda,
                                       const float* __restrict__ Bsw, int K,
                                       int am, int lane, v8f& acc) {
  const int hi = lane >> 4;
  const float* arow = A + am * lda + 2 * hi;
  const float* bp = Bsw + lane * 2;
#pragma unroll 4
  for (int k0 = 0; k0 < K; k0 += 4) {
    v2f a = *(const v2f*)(arow + k0);
    v2f b = *(const v2f*)(bp + (k0 << 4));
    acc = wmma4(a, b, acc);
  }
}

// ---------------------------------------------------------------------------
// One Jacobi sweep: fused 4-layer update. 784 wave-jobs:
//   3*256 strips (u1',u2',u3': 16 m-tiles x 16 strip-columns) + 16 tiles (u4').
// ---------------------------------------------------------------------------
__global__ __launch_bounds__(128) void eq_iter(
    const float* __restrict__ U1i, const float* __restrict__ U2i,
    const float* __restrict__ U3i, const float* __restrict__ U4i,
    float* __restrict__ U1o, float* __restrict__ U2o,
    float* __restrict__ U3o, float* __restrict__ U4o,
    const float* __restrict__ W1sw,  const float* __restrict__ W1Tsw,
    const float* __restrict__ W2sw,  const float* __restrict__ W2Tsw,
    const float* __restrict__ W3sw,  const float* __restrict__ W3Tsw,
    const float* __restrict__ xW0,   const float* __restrict__ b1p,
    const float* __restrict__ b2p,   const float* __restrict__ b3p,
    const float* __restrict__ tpad,  float beta) {
  const int wave = blockIdx.x * 4 + (threadIdx.x >> 5);  // uniform per wave
  const int lane = threadIdx.x & 31;
  const int hi = lane >> 4, lo = lane & 15;

  if (wave < 768) {
    const int mat = wave >> 8;      // 0: u1', 1: u2', 2: u3'
    const int job = wave & 255;
    const int m0 = (job >> 4) * 16;
    const int nt0 = (job & 15) * 2;
    const int n0 = nt0 * 16;
    const int am = m0 + lo;

    v8f acc0 = {}, acc1 = {};
    if (mat == 0) {                 // pre1 = xW0 + u2 @ W1^T
      gemm2s(U2i, LDU, W1Tsw, nt0, 8192, LDU, am, lane, acc0, acc1);
    } else if (mat == 1) {          // pre2 = u1 @ W1 + u3 @ W2^T + b1
      gemm2s(U1i, LDU, W1sw,  nt0, 8192, LDU, am, lane, acc0, acc1);
      gemm2s(U3i, LDU, W2Tsw, nt0, 8192, LDU, am, lane, acc0, acc1);
    } else {                        // pre3 = u2 @ W2 + u4 @ W3^T + b2
      gemm2s(U2i, LDU,  W2sw,  nt0, 8192, LDU,  am, lane, acc0, acc1);
      gemm2s(U4i, LDU4, W3Tsw, nt0, 256,  LDU4, am, lane, acc0, acc1);
    }

    const float* Ui = (mat == 0) ? U1i : (mat == 1) ? U2i : U3i;
    float*       Uo = (mat == 0) ? U1o : (mat == 1) ? U2o : U3o;
    const int c0 = n0 + lo, c1 = n0 + 16 + lo;
#pragma unroll
    for (int i = 0; i < 8; ++i) {
      const int m = m0 + i + 8 * hi;
      float add0, add1;
      if (mat == 0) { const float* xr = xW0 + m * LDU; add0 = xr[c0]; add1 = xr[c1]; }
      else          { const float* bp = (mat == 1) ? b1p : b2p; add0 = bp[c0]; add1 = bp[c1]; }
      const float* ur = Ui + m * LDU;
      float*       orow = Uo + m * LDU;
      float u0 = ur[c0], u1v = ur[c1];
      float pre0 = acc0[i] + add0;
      float pre1 = acc1[i] + add1;
      float dr0 = (u0  >= 0.f && u0  <= 1.f) ? 1.f : 0.f;
      float dr1 = (u1v >= 0.f && u1v <= 1.f) ? 1.f : 0.f;
      float d0 = fmaf(dr0, pre0, -u0);
      float d1 = fmaf(dr1, pre1, -u1v);
      orow[c0] = fminf(fmaxf(fmaf(EPS, d0, u0),  0.f), 1.f);
      orow[c1] = fminf(fmaxf(fmaf(EPS, d1, u1v), 0.f), 1.f);
    }
  } else {                          // u4': pre4 = u3 @ W3 + b3 (+ beta*(t-u4))
    const int m0 = (wave - 768) * 16;
    v8f acc = {};
    gemm1s(U3i, LDU, W3sw, LDU, m0 + lo, lane, acc);
#pragma unroll
    for (int i = 0; i < 8; ++i) {
      const int m = m0 + i + 8 * hi;
      float pre = acc[i] + b3p[lo];
      float u = U4i[m * LDU4 + lo];
      float dr = (u >= 0.f && u <= 1.f) ? 1.f : 0.f;
      float d = fmaf(dr, pre, -u) + beta * (tpad[m * LDU4 + lo] - u);
      U4o[m * LDU4 + lo] = fminf(fmaxf(fmaf(EPS, d, u), 0.f), 1.f);
    }
  }
}

// ---------------------------------------------------------------------------
// xW0 = x @ W0 + b0 via WMMA (K = 784 = 196 k-chunks), run once per replay.
// 256 wave-jobs: 16 m-tiles x 16 strip-columns.
// ---------------------------------------------------------------------------
__global__ __launch_bounds__(128) void xw0_wmma(const float* __restrict__ x,
                                                const float* __restrict__ W0sw,
                                                const float* __restrict__ b0p,
                                                float* __restrict__ xW0p) {
  const int wave = blockIdx.x * 4 + (threadIdx.x >> 5);
  const int lane = threadIdx.x & 31;
  const int hi = lane >> 4, lo = lane & 15;
  const int m0 = (wave >> 4) * 16;
  const int nt0 = (wave & 15) * 2;
  v8f acc0 = {}, acc1 = {};
  gemm2s(x, 784, W0sw, nt0, 12544, 784, m0 + lo, lane, acc0, acc1);
  const int c0 = nt0 * 16 + lo, c1 = c0 + 16;
#pragma unroll
  for (int i = 0; i < 8; ++i) {
    const int m = m0 + i + 8 * hi;
    xW0p[m * LDU + c0] = acc0[i] + b0p[c0];
    xW0p[m * LDU + c1] = acc1[i] + b0p[c1];
  }
}

// ---------------------------------------------------------------------------
// One-time setup: swizzle weights into WMMA fragment order, pad, init state.
// ---------------------------------------------------------------------------
__global__ void pad_w500_swz(const float* __restrict__ W, float* __restrict__ Wsw,
                             float* __restrict__ WTsw) {
  int idx = blockIdx.x * 256 + threadIdx.x;   // 32 nt * 128 kc * 64 = 262144
  int nt = idx >> 13, kc = (idx >> 6) & 127, lane = (idx >> 1) & 31, j = idx & 1;
  int k = kc * 4 + 2 * (lane >> 4) + j;
  int n = nt * 16 + (lane & 15);
  bool in = (k < 500) && (n < 500);
  Wsw[idx]  = in ? W[k * 500 + n] : 0.f;      // B = W
  WTsw[idx] = in ? W[n * 500 + k] : 0.f;      // B = W^T
}

__global__ void pad_w0_swz(const float* __restrict__ W0, float* __restrict__ W0sw) {
  int idx = blockIdx.x * 256 + threadIdx.x;   // 32 nt * 196 kc * 64 = 401408
  int nt = idx / 12544, r = idx % 12544;
  int kc = r >> 6, lane = (r >> 1) & 31, j = r & 1;
  int k = kc * 4 + 2 * (lane >> 4) + j;       // < 784 always
  int n = nt * 16 + (lane & 15);
  W0sw[idx] = (n < 500) ? W0[k * 500 + n] : 0.f;
}

__global__ void pad_w3_swz(const float* __restrict__ W3, const float* __restrict__ b3,
                           float* __restrict__ W3sw, float* __restrict__ W3Tsw,
                           float* __restrict__ b3p) {
  int idx = blockIdx.x * 256 + threadIdx.x;   // 8192 + 8192
  if (idx < 8192) {                           // W3sw: 1 nt * 128 kc * 64
    int kc = idx >> 6, lane = (idx >> 1) & 31, j = idx & 1;
    int k = kc * 4 + 2 * (lane >> 4) + j, n = lane & 15;
    W3sw[idx] = (k < 500 && n < 10) ? W3[k * 10 + n] : 0.f;
  } else {                                    // W3Tsw: 32 nt * 4 kc * 64
    int r = idx - 8192;
    int nt = r >> 8, rr = r & 255;
    int kc = rr >> 6, lane = (rr >> 1) & 31, j = rr & 1;
    int k = kc * 4 + 2 * (lane >> 4) + j;
    int n = nt * 16 + (lane & 15);
    W3Tsw[r] = (k < 10 && n < 500) ? W3[n * 10 + k] : 0.f;
  }
  if (idx < 16) b3p[idx] = (idx < 10) ? b3[idx] : 0.f;
}

__global__ void pad_bias3(const float* __restrict__ b0, const float* __restrict__ b1,
                          const float* __restrict__ b2, float* __restrict__ b0p,
                          float* __restrict__ b1p, float* __restrict__ b2p) {
  int idx = blockIdx.x * 256 + threadIdx.x;   // 1536
  int which = idx >> 9, n = idx & 511;
  const float* s = (which == 0) ? b0 : (which == 1) ? b1 : b2;
  float*       d = (which == 0) ? b0p : (which == 1) ? b1p : b2p;
  d[n] = (n < 500) ? s[n] : 0.f;
}

__global__ void pad_t(const float* __restrict__ t, float* __restrict__ tp) {
  int idx = blockIdx.x * 256 + threadIdx.x;   // 256*16
  int b = idx >> 4, n = idx & 15;
  tp[idx] = (n < 10) ? t[b * 10 + n] : 0.f;
}

__global__ void init_u(const float* __restrict__ u1, const float* __restrict__ u2,
                       const float* __restrict__ u3, const float* __restrict__ u4,
                       float* __restrict__ U1, float* __restrict__ U2,
                       float* __restrict__ U3, float* __restrict__ U4) {
  int idx = blockIdx.x * 256 + threadIdx.x;   // 3*256*512 + 256*16
  if (idx < 393216) {
    int which = idx / 131072, r = idx % 131072;
    int b = r >> 9, n = r & 511;
    const float* src = (which == 0) ? u1 : (which == 1) ? u2 : u3;
    float* dst = (which == 0) ? U1 : (which == 1) ? U2 : U3;
    dst[r] = (n < 500) ? src[b * 500 + n] : 0.f;
  } else {
    int r = idx - 393216;                     // 0..4095
    int b = r >> 4, n = r & 15;
    U4[r] = (n < 10) ? u4[b * 10 + n] : 0.f;
  }
}

__global__ void export_states(const float* __restrict__ U1, const float* __restrict__ U2,
                              const float* __restrict__ U3, const float* __restrict__ U4,
                              float* __restrict__ out) {
  int idx = blockIdx.x * 256 + threadIdx.x;   // 386560
  if (idx >= 386560) return;
  if (idx < 384000) {
    int which = idx / 128000, r = idx % 128000;
    int b = r / 500, n = r % 500;
    const float* U = (which == 0) ? U1 : (which == 1) ? U2 : U3;
    out[idx] = U[b * LDU + n];
  } else {
    int r = idx - 384000;
    int b = r / 10, n = r % 10;
    out[idx] = U4[b * LDU4 + n];
  }
}

// ---------------------------------------------------------------------------
extern "C" void kernel_launch(void* const* d_in, const int* in_sizes, int n_in,
                              void* d_out, int out_size, void* d_ws, size_t ws_size,
                              hipStream_t stream) {
  const float* x  = (const float*)d_in[0];
  const float* u1 = (const float*)d_in[1];
  const float* u2 = (const float*)d_in[2];
  const float* u3 = (const float*)d_in[3];
  const float* u4 = (const float*)d_in[4];
  const float* t  = (const float*)d_in[5];
  const float* W0 = (const float*)d_in[6];
  const float* W1 = (const float*)d_in[7];
  const float* W2 = (const float*)d_in[8];
  const float* W3 = (const float*)d_in[9];
  const float* b0 = (const float*)d_in[10];
  const float* b1 = (const float*)d_in[11];
  const float* b2 = (const float*)d_in[12];
  const float* b3 = (const float*)d_in[13];

  float* ws = (float*)d_ws;
  float* W1sw  = ws;                  // 262144
  float* W1Tsw = W1sw  + 262144;      // 262144
  float* W2sw  = W1Tsw + 262144;      // 262144
  float* W2Tsw = W2sw  + 262144;      // 262144
  float* W3sw  = W2Tsw + 262144;      // 8192
  float* W3Tsw = W3sw  + 8192;        // 8192
  float* W0sw  = W3Tsw + 8192;        // 401408
  float* b0p   = W0sw  + 401408;      // 512
  float* b1p   = b0p   + 512;         // 512
  float* b2p   = b1p   + 512;         // 512
  float* b3p   = b2p   + 512;         // 16
  float* xW0p  = b3p   + 16;          // 131072
  float* tpad  = xW0p  + 131072;      // 4096
  float* base  = tpad  + 4096;
  float* U1b[2] = { base,              base + 131072 };
  float* U2b[2] = { base + 2 * 131072, base + 3 * 131072 };
  float* U3b[2] = { base + 4 * 131072, base + 5 * 131072 };
  float* U4b[2] = { base + 6 * 131072, base + 6 * 131072 + 4096 };

  pad_w500_swz<<<1024, 256, 0, stream>>>(W1, W1sw, W1Tsw);
  pad_w500_swz<<<1024, 256, 0, stream>>>(W2, W2sw, W2Tsw);
  pad_w3_swz  <<<64,   256, 0, stream>>>(W3, b3, W3sw, W3Tsw, b3p);
  pad_w0_swz  <<<1568, 256, 0, stream>>>(W0, W0sw);
  pad_bias3   <<<6,    256, 0, stream>>>(b0, b1, b2, b0p, b1p, b2p);
  pad_t       <<<16,   256, 0, stream>>>(t, tpad);
  init_u      <<<1552, 256, 0, stream>>>(u1, u2, u3, u4,
                                         U1b[0], U2b[0], U3b[0], U4b[0]);
  xw0_wmma    <<<64,   128, 0, stream>>>(x, W0sw, b0p, xW0p);

  // 500 free sweeps (Jacobi barrier = kernel boundary)
  for (int it = 0; it < 500; ++it) {
    int s = it & 1, d = s ^ 1;
    eq_iter<<<196, 128, 0, stream>>>(U1b[s], U2b[s], U3b[s], U4b[s],
                                     U1b[d], U2b[d], U3b[d], U4b[d],
                                     W1sw, W1Tsw, W2sw, W2Tsw, W3sw, W3Tsw,
                                     xW0p, b1p, b2p, b3p, tpad, 0.0f);
  }
  // after an even number of sweeps the state is back in buffer 0
  export_states<<<1510, 256, 0, stream>>>(U1b[0], U2b[0], U3b[0], U4b[0],
                                          (float*)d_out);
  // 8 clamped sweeps (beta = 1)
  for (int it = 0; it < 8; ++it) {
    int s = it & 1, d = s ^ 1;
    eq_iter<<<196, 128, 0, stream>>>(U1b[s], U2b[s], U3b[s], U4b[s],
                                     U1b[d], U2b[d], U3b[d], U4b[d],
                                     W1sw, W1Tsw, W2sw, W2Tsw, W3sw, W3Tsw,
                                     xW0p, b1p, b2p, b3p, tpad, 1.0f);
  }
  export_states<<<1510, 256, 0, stream>>>(U1b[0], U2b[0], U3b[0], U4b[0],
                                          (float*)d_out + 386560);
}